// LightGCN_MACR_50337016709122
// MI455X (gfx1250) — compile-verified
//
#include <hip/hip_runtime.h>
#include <hip/hip_bf16.h>

#define DIM 64

typedef float v2f __attribute__((ext_vector_type(2)));
typedef float v8f __attribute__((ext_vector_type(8)));

// ---------------------------------------------------------------------------
// Self-gating: G = X @ W + b ; gate = X * sigmoid(G)
// One wave32 per 16x16 output tile, K=64 via 16 chained V_WMMA_F32_16X16X4_F32.
// Writes raw embedding into acc/cur and gated embedding into accg/curg.
// ---------------------------------------------------------------------------
__global__ void __launch_bounds__(256)
gating_kernel(const float* __restrict__ demb, const float* __restrict__ remb,
              const float* __restrict__ wd,  const float* __restrict__ bd,
              const float* __restrict__ wr,  const float* __restrict__ br,
              float* __restrict__ acc,  float* __restrict__ accg,
              float* __restrict__ cur,  float* __restrict__ curg,
              int n_d, int n_r)
{
    int wave = (int)((blockIdx.x * blockDim.x + threadIdx.x) >> 5);
    int lane = threadIdx.x & 31;
    int rows = n_d + n_r;
    int n_tiles = (rows >> 4) * (DIM / 16);      // row tiles x 4 col tiles
    if (wave >= n_tiles) return;                 // wave-uniform branch (EXEC all-1 inside)

    int row_tile = wave >> 2;
    int col_tile = wave & 3;
    int r0 = row_tile << 4;
    int n0 = col_tile << 4;

    const float* X; const float* W; const float* bias; int rbase;
    if (r0 < n_d) { X = demb; W = wd; bias = bd; rbase = r0; }
    else          { X = remb; W = wr; bias = br; rbase = r0 - n_d; }

    int ln   = lane & 15;     // A: row M ; B: col N ; C: col N
    int half = lane >> 4;     // selects K pair (A/B) and M half (C/D)

    v8f c = {0.f, 0.f, 0.f, 0.f, 0.f, 0.f, 0.f, 0.f};
    #pragma unroll
    for (int k = 0; k < DIM; k += 4) {
        int ka = k + 2 * half;
        v2f a, b;
        // A 16x4 f32 layout: lane m (0-15): VGPR0=K(2*half), VGPR1=K(2*half+1)
        a.x = X[(rbase + ln) * DIM + ka];
        a.y = X[(rbase + ln) * DIM + ka + 1];
        // B 4x16 f32 layout: row K striped across lanes within a VGPR
        b.x = W[(ka)     * DIM + n0 + ln];
        b.y = W[(ka + 1) * DIM + n0 + ln];
        c = __builtin_amdgcn_wmma_f32_16x16x4_f32(false, a, false, b,
                                                  (short)0, c, false, false);
    }

    #pragma unroll
    for (int v = 0; v < 8; ++v) {
        int m   = v + 8 * half;            // C/D layout: VGPR v -> M = v + 8*half
        int row = r0 + m;
        int col = n0 + ln;
        float g    = c[v] + bias[col];
        float x    = X[(rbase + m) * DIM + col];
        float gate = x * (1.f / (1.f + __expf(-g)));
        size_t o = (size_t)row * DIM + col;
        acc[o]  = x;    cur[o]  = x;
        accg[o] = gate; curg[o] = gate;
    }
}

// ---------------------------------------------------------------------------
// Utility kernels
// ---------------------------------------------------------------------------
__global__ void zero_kernel(float* __restrict__ p, long n)
{
    long i      = (long)blockIdx.x * blockDim.x + threadIdx.x;
    long stride = (long)gridDim.x * blockDim.x;
    for (; i < n; i += stride) p[i] = 0.f;
}

__global__ void zero_scalar_kernel(float* p) { if (threadIdx.x == 0) p[0] = 0.f; }

__global__ void add_kernel(float* __restrict__ acc, const float* __restrict__ x, long n)
{
    long i      = (long)blockIdx.x * blockDim.x + threadIdx.x;
    long stride = (long)gridDim.x * blockDim.x;
    for (; i < n; i += stride) acc[i] += x[i];
}

// ---------------------------------------------------------------------------
// Sparse propagation: next[dst] += cur[src] * val   (segment_sum via atomics)
// 16 lanes per edge, float4 per lane -> coalesced 256B row gather/scatter.
// ---------------------------------------------------------------------------
__global__ void __launch_bounds__(256)
prop_kernel(const int* __restrict__ idx, const float* __restrict__ val,
            const float* __restrict__ cur, float* __restrict__ next, int n_e)
{
    long t = (long)blockIdx.x * blockDim.x + threadIdx.x;
    long e = t >> 4;
    if (e >= n_e) return;
    int c4  = (int)(t & 15);
    int dst = idx[e];
    int src = idx[(long)n_e + e];
    float v = val[e];
    const float4 x = *(const float4*)(cur + (size_t)src * DIM + c4 * 4);
    float* p = next + (size_t)dst * DIM + c4 * 4;
    atomicAdd(p + 0, x.x * v);
    atomicAdd(p + 1, x.y * v);
    atomicAdd(p + 2, x.z * v);
    atomicAdd(p + 3, x.w * v);
}

// ---------------------------------------------------------------------------
// Fuse + score + weighted BCE.  fuse = 0.5*(acc/4) + 0.5*(accg/4) = (acc+accg)/8
// out[0] = loss (atomic mean), out[1..B] = sigmoid(scores)
// ---------------------------------------------------------------------------
__global__ void __launch_bounds__(256)
final_kernel(const float* __restrict__ acc, const float* __restrict__ accg,
             const int* __restrict__ dis, const int* __restrict__ drg,
             const int* __restrict__ lab, float* __restrict__ out,
             int n_d, int n_b)
{
    int i = blockIdx.x * blockDim.x + threadIdx.x;
    if (i >= n_b) return;
    size_t dr = (size_t)dis[i] * DIM;
    size_t rr = ((size_t)n_d + drg[i]) * DIM;

    float score = 0.f, sbd = 0.f, sbr = 0.f;
    #pragma unroll 4
    for (int j = 0; j < DIM; ++j) {
        float bdj = (acc[dr + j] + accg[dr + j]) * 0.125f;
        float brj = (acc[rr + j] + accg[rr + j]) * 0.125f;
        score += bdj * brj;
        sbd   += 1.f / (1.f + __expf(-bdj));
        sbr   += 1.f / (1.f + __expf(-brj));
    }
    float sbias = score * sbd * sbr;
    float p  = 1.f / (1.f + __expf(-score));
    float pb = 1.f / (1.f + __expf(-sbias));

    float l = (float)lab[i];
    float w = 5.0f * l + 1.0f - l;                 // POS_W = 5
    float lp   = fmaxf(logf(p),    -100.f);
    float l1p  = fmaxf(log1pf(-p), -100.f);
    float loss1 = w * -(l * lp + (1.f - l) * l1p);
    float lpb  = fmaxf(logf(pb),    -100.f);
    float l1pb = fmaxf(log1pf(-pb), -100.f);
    float loss2 = w * -(l * lpb + (1.f - l) * l1pb);

    out[1 + i] = p;
    atomicAdd(out, (loss1 + 0.1f * loss2) / (float)n_b);
}

// ---------------------------------------------------------------------------
extern "C" void kernel_launch(void* const* d_in, const int* in_sizes, int n_in,
                              void* d_out, int out_size, void* d_ws, size_t ws_size,
                              hipStream_t stream)
{
    const float* demb = (const float*)d_in[0];
    const float* remb = (const float*)d_in[1];
    const float* gw_d  = (const float*)d_in[2];
    const float* gw_db = (const float*)d_in[3];
    const float* gw_r  = (const float*)d_in[4];
    const float* gw_rb = (const float*)d_in[5];
    const int*   eidx1 = (const int*)d_in[6];
    const float* eval1 = (const float*)d_in[7];
    const int*   eidx2 = (const int*)d_in[8];
    const float* eval2 = (const float*)d_in[9];
    const int*   dis   = (const int*)d_in[10];
    const int*   drg   = (const int*)d_in[11];
    const int*   lab   = (const int*)d_in[12];

    const int n_d = in_sizes[0] / DIM;      // 20000
    const int n_r = in_sizes[1] / DIM;      // 20000
    const int n_e = in_sizes[7];            // 1280000
    const int n_b = in_sizes[10];           // 8192
    const int rows = n_d + n_r;
    const long S = (long)rows * DIM;        // floats per node-feature buffer

    float* ws   = (float*)d_ws;
    float* acc   = ws;
    float* accg  = ws + 1 * S;
    float* bufA  = ws + 2 * S;
    float* bufB  = ws + 3 * S;
    float* bufgA = ws + 4 * S;
    float* bufgB = ws + 5 * S;
    float* out = (float*)d_out;

    // --- gating + init acc/cur (exact-f32 WMMA GEMM) ---
    int n_tiles = (rows >> 4) * (DIM / 16);               // 10000 waves
    int gblocks = (n_tiles * 32 + 255) / 256;             // 1250 blocks
    gating_kernel<<<gblocks, 256, 0, stream>>>(demb, remb, gw_d, gw_db, gw_r, gw_rb,
                                               acc, accg, bufA, bufgA, n_d, n_r);
    zero_scalar_kernel<<<1, 64, 0, stream>>>(out);

    // --- 3 propagation layers, ping-pong bufA<->bufB (and gated twins) ---
    const int pblocks = (int)(((long)n_e * 16 + 255) / 256);
    const int zblocks = 2048, ablocks = 2048;

    float* cur  = bufA; float* nxt  = bufB;
    float* curg = bufgA; float* nxtg = bufgB;
    for (int layer = 0; layer < 3; ++layer) {
        zero_kernel<<<zblocks, 256, 0, stream>>>(nxt,  S);
        zero_kernel<<<zblocks, 256, 0, stream>>>(nxtg, S);
        prop_kernel<<<pblocks, 256, 0, stream>>>(eidx1, eval1, cur,  nxt,  n_e);
        prop_kernel<<<pblocks, 256, 0, stream>>>(eidx2, eval2, curg, nxtg, n_e);
        add_kernel<<<ablocks, 256, 0, stream>>>(acc,  nxt,  S);
        add_kernel<<<ablocks, 256, 0, stream>>>(accg, nxtg, S);
        float* t;
        t = cur;  cur  = nxt;  nxt  = t;
        t = curg; curg = nxtg; nxtg = t;
    }

    // --- fuse + scores + loss ---
    final_kernel<<<(n_b + 255) / 256, 256, 0, stream>>>(acc, accg, dis, drg, lab,
                                                        out, n_d, n_b);
}